// PureMul32_62380105007299
// MI455X (gfx1250) — compile-verified
//
#include <hip/hip_runtime.h>

// ---------------------------------------------------------------------------
// PureMul32 circuit chain for gfx1250 (MI455X), f32 WMMA (V_WMMA_F32_16X16X4_F32)
// 4 circuits = two 16x16 tiles per wave32 (B operands + barriers amortized).
// Per-iteration weights staged (pre-padded) into LDS once per workgroup.
// ---------------------------------------------------------------------------

typedef float v2f __attribute__((ext_vector_type(2)));
typedef float v8f __attribute__((ext_vector_type(8)));

#define WMMA4(A, B, C) \
  __builtin_amdgcn_wmma_f32_16x16x4_f32(false, (A), false, (B), (short)0, (C), false, false)

#define WSTR 18   // weight row stride in LDS: banks (18n+k)%64 conflict-free, 8B aligned
#define CSTR 18   // tile scratch column stride (col-major staging)

// ---------------- global-memory B operand (prologue only) -------------------
template <int ROWS, int COLS, bool GN, bool GK>
__device__ __forceinline__ v2f load_b(const float* __restrict__ W, int n_off, int j, int lane) {
  const int n = n_off + (lane & 15);
  const int k = 4 * j + 2 * (lane >> 4);
  const int nc = GN ? (n & (ROWS - 1)) : n;
  const int kc = GK ? (k & (COLS - 1)) : k;
  const float t0 = W[nc * COLS + kc + 0];
  const float t1 = W[nc * COLS + kc + 1];
  const bool pn  = !GN || (n < ROWS);
  const bool pk0 = !GK || (k + 0 < COLS);
  const bool pk1 = !GK || (k + 1 < COLS);
  v2f b;
  b.x = (pn && pk0) ? t0 : 0.0f;
  b.y = (pn && pk1) ? t1 : 0.0f;
  return b;
}

template <int H>
__device__ __forceinline__ float load_bias(const float* __restrict__ b, int n) {
  if (H >= 16) return b[n];
  const float t = b[n & (H - 1)];
  return (n < H) ? t : 0.0f;
}

// ---------------- LDS-staged B operand (hot loop) ---------------------------
__device__ __forceinline__ v2f lds_b(const float* w, int j, int lane) {
  const int n = lane & 15;
  const int k = 4 * j + 2 * (lane >> 4);
  const float* p = w + n * WSTR + k;     // even word index -> 8B aligned
  v2f b;
  b.x = p[0];
  b.y = p[1];
  return b;
}

// ---------------- tile staging (col-major, stride CSTR) ---------------------
__device__ __forceinline__ v2f lds_a(const float* sc, int j, int lane) {
  const int row = lane & 15;
  const int c0  = 4 * j + 2 * (lane >> 4);
  v2f a;
  a.x = sc[(c0 + 0) * CSTR + row];
  a.y = sc[(c0 + 1) * CSTR + row];
  return a;
}

__device__ __forceinline__ void stage_tile(float* sc, const v8f& T, int lane, int col_off) {
  const int col = (lane & 15) + col_off;
  const int r0  = 8 * (lane >> 4);
  float* p = sc + col * CSTR + r0;       // even word index -> 8B aligned
#pragma unroll
  for (int q = 0; q < 4; ++q) {
    v2f t;
    t.x = T[2 * q + 0];
    t.y = T[2 * q + 1];
    *reinterpret_cast<v2f*>(p + 2 * q) = t;
  }
}

__device__ __forceinline__ v8f gated(const v8f& up, float vbu, const v8f& gt, float vbg) {
  v8f h;
#pragma unroll
  for (int r = 0; r < 8; ++r) {
    float u = up[r] + vbu;
    u = fmaxf(u, 0.0f);
    h[r] = u * (gt[r] + vbg);
  }
  return h;
}

// ---------------- FFN pair with LDS-staged (pre-padded) weights -------------
template <int KC>   // KC = ceil(H/4) chunks for the down-projection
__device__ __forceinline__ void ffn_pair(v8f& Xa, v8f& Xb, float* sca, float* scb, int lane,
                                         const float* wU, const float* vb_u,
                                         const float* wG, const float* vb_g,
                                         const float* wD) {
  const int n = lane & 15;

  stage_tile(sca, Xa, lane, 0);
  stage_tile(scb, Xb, lane, 0);
  v2f Aa[4], Ab[4];
#pragma unroll
  for (int j = 0; j < 4; ++j) { Aa[j] = lds_a(sca, j, lane); Ab[j] = lds_a(scb, j, lane); }

  v8f ua = {0, 0, 0, 0, 0, 0, 0, 0}, ga = ua, ub = ua, gb = ua;
#pragma unroll
  for (int j = 0; j < 4; ++j) {          // B operands shared by both tiles
    const v2f Bu = lds_b(wU, j, lane);
    const v2f Bg = lds_b(wG, j, lane);
    ua = WMMA4(Aa[j], Bu, ua);
    ub = WMMA4(Ab[j], Bu, ub);
    ga = WMMA4(Aa[j], Bg, ga);
    gb = WMMA4(Ab[j], Bg, gb);
  }
  const float vbu = vb_u[n];             // padded: cols >= H are exactly 0
  const float vbg = vb_g[n];

  stage_tile(sca, gated(ua, vbu, ga, vbg), lane, 0);
  stage_tile(scb, gated(ub, vbu, gb, vbg), lane, 0);

#pragma unroll
  for (int j = 0; j < KC; ++j) {
    const v2f Bd = lds_b(wD, j, lane);
    Xa = WMMA4(lds_a(sca, j, lane), Bd, Xa);
    Xb = WMMA4(lds_a(scb, j, lane), Bd, Xb);
  }
}

// ---------------- init FFN pair (H=2, one-time, weights from global) --------
__device__ __forceinline__ void ffn_init_pair(v8f& Xa, v8f& Xb, float* sca, float* scb, int lane,
                                              const float* __restrict__ Wu,
                                              const float* __restrict__ bu,
                                              const float* __restrict__ Wg,
                                              const float* __restrict__ bg,
                                              const float* __restrict__ Wd) {
  const int n = lane & 15;
  stage_tile(sca, Xa, lane, 0);
  stage_tile(scb, Xb, lane, 0);
  v2f Aa[4], Ab[4];
#pragma unroll
  for (int j = 0; j < 4; ++j) { Aa[j] = lds_a(sca, j, lane); Ab[j] = lds_a(scb, j, lane); }
  v8f ua = {0, 0, 0, 0, 0, 0, 0, 0}, ga = ua, ub = ua, gb = ua;
#pragma unroll
  for (int j = 0; j < 4; ++j) {
    const v2f Bu = load_b<2, 16, true, false>(Wu, 0, j, lane);
    const v2f Bg = load_b<2, 16, true, false>(Wg, 0, j, lane);
    ua = WMMA4(Aa[j], Bu, ua);
    ub = WMMA4(Ab[j], Bu, ub);
    ga = WMMA4(Aa[j], Bg, ga);
    gb = WMMA4(Ab[j], Bg, gb);
  }
  const float vbu = load_bias<2>(bu, n);
  const float vbg = load_bias<2>(bg, n);
  stage_tile(sca, gated(ua, vbu, ga, vbg), lane, 0);
  stage_tile(scb, gated(ub, vbu, gb, vbg), lane, 0);
  const v2f Bd = load_b<16, 2, false, true>(Wd, 0, 0, lane);
  Xa = WMMA4(lds_a(sca, 0, lane), Bd, Xa);
  Xb = WMMA4(lds_a(scb, 0, lane), Bd, Xb);
}

// ---------------- carry FFN pair (H=32, weights hoisted in VGPRs) -----------
struct CarryW {
  v2f Bu[2][4];
  v2f Bg[2][4];
  v2f Bd[8];
  float bu[2], bg[2];
};

__device__ __forceinline__ void carry_pair(v8f& Xa, v8f& Xb, float* sca, float* scb,
                                           int lane, const CarryW& w) {
  stage_tile(sca, Xa, lane, 0);
  stage_tile(scb, Xb, lane, 0);
  v2f Aa[4], Ab[4];
#pragma unroll
  for (int j = 0; j < 4; ++j) { Aa[j] = lds_a(sca, j, lane); Ab[j] = lds_a(scb, j, lane); }

  // N=32 processed half-by-half to cap live accumulators.
#pragma unroll
  for (int hh = 0; hh < 2; ++hh) {
    v8f ua = {0, 0, 0, 0, 0, 0, 0, 0}, ga = ua, ub = ua, gb = ua;
#pragma unroll
    for (int j = 0; j < 4; ++j) {
      ua = WMMA4(Aa[j], w.Bu[hh][j], ua);
      ub = WMMA4(Ab[j], w.Bu[hh][j], ub);
      ga = WMMA4(Aa[j], w.Bg[hh][j], ga);
      gb = WMMA4(Ab[j], w.Bg[hh][j], gb);
    }
    stage_tile(sca, gated(ua, w.bu[hh], ga, w.bg[hh]), lane, 16 * hh);
    stage_tile(scb, gated(ub, w.bu[hh], gb, w.bg[hh]), lane, 16 * hh);
  }
#pragma unroll
  for (int j = 0; j < 8; ++j) {          // K=32 down-projection, C = X
    Xa = WMMA4(lds_a(sca, j, lane), w.Bd[j], Xa);
    Xb = WMMA4(lds_a(scb, j, lane), w.Bd[j], Xb);
  }
}

// ---------------------------------------------------------------------------
__global__ __launch_bounds__(256) void pure_mul32_kernel(
    const float* __restrict__ x,
    const float* __restrict__ Wu0, const float* __restrict__ bu0,
    const float* __restrict__ Wg0, const float* __restrict__ bg0,
    const float* __restrict__ Wd0,
    const float* __restrict__ Wu_e, const float* __restrict__ bu_e,
    const float* __restrict__ Wg_e, const float* __restrict__ bg_e,
    const float* __restrict__ Wd_e,
    const float* __restrict__ Wu_s, const float* __restrict__ bu_s,
    const float* __restrict__ Wg_s, const float* __restrict__ bg_s,
    const float* __restrict__ Wd_s,
    const float* __restrict__ Wu_c, const float* __restrict__ bu_c,
    const float* __restrict__ Wg_c, const float* __restrict__ bg_c,
    const float* __restrict__ Wd_c,
    float* __restrict__ out, int npairs) {
  // Per-workgroup weight staging (pre-padded) + per-wave tile scratch (x2).
  __shared__ float eU[16 * WSTR], eG[16 * WSTR], eD[16 * WSTR];   // extract (H=4)
  __shared__ float sU[16 * WSTR], sG[16 * WSTR], sD[16 * WSTR];   // shifted (H=16)
  __shared__ float ebu[16], ebg[16], sbu[16], sbg[16];
  __shared__ float scratch[8][2][32 * CSTR];

  const int t    = threadIdx.x;
  const int wave = t >> 5;
  const int lane = t & 31;
  float* sca = &scratch[wave][0][0];
  float* scb = &scratch[wave][1][0];

  int pair = blockIdx.x * 8 + wave;            // uniform per wave
  if (pair >= npairs) pair = npairs - 1;       // clamp: keep all waves in barriers

  const int n  = lane & 15;
  const int r0 = 8 * (lane >> 4);

  // Zero the static pad regions once (extract rows 4..15 and bias tails).
  for (int idx = t; idx < 16 * WSTR; idx += 256) { eU[idx] = 0.0f; eG[idx] = 0.0f; }
  if (t < 16) { ebu[t] = 0.0f; ebg[t] = 0.0f; }

  // Load 2 tiles (4 circuits = contiguous 512 floats) into C/D layout.
  const float* xin = x + (size_t)pair * 512;
  v8f Xa, Xb;
#pragma unroll
  for (int r = 0; r < 8; ++r) {
    Xa[r] = xin[(r0 + r) * 16 + n];
    Xb[r] = xin[256 + (r0 + r) * 16 + n];
  }

  // Hoist iteration-invariant carry-FFN weights into VGPRs (all in-range).
  CarryW cw;
#pragma unroll
  for (int hh = 0; hh < 2; ++hh) {
#pragma unroll
    for (int j = 0; j < 4; ++j) {
      cw.Bu[hh][j] = load_b<32, 16, false, false>(Wu_c, 16 * hh, j, lane);
      cw.Bg[hh][j] = load_b<32, 16, false, false>(Wg_c, 16 * hh, j, lane);
    }
    cw.bu[hh] = bu_c[16 * hh + n];
    cw.bg[hh] = bg_c[16 * hh + n];
  }
#pragma unroll
  for (int j = 0; j < 8; ++j) cw.Bd[j] = load_b<16, 32, false, false>(Wd_c, 0, j, lane);

  // Init accumulator FFN (H=2), weights straight from global (one-time).
  ffn_init_pair(Xa, Xb, sca, scb, lane, Wu0, bu0, Wg0, bg0, Wd0);

  // 32 iterations: extract-bit (H=4), shifted-add (H=16), carry (H=32).
#pragma unroll 1
  for (int i = 0; i < 32; ++i) {
    // ---- cooperative, coalesced weight staging for this iteration ----
    {
      const int eb = i * 64, sb = i * 256;
      if (t < 64) {                            // extract: 4x16 blocks
        eU[(t >> 4) * WSTR + (t & 15)] = Wu_e[eb + t];
        eG[(t >> 4) * WSTR + (t & 15)] = Wg_e[eb + t];
        eD[(t >> 2) * WSTR + (t & 3)]  = Wd_e[eb + t];   // 16x4 block
      }
      sU[(t >> 4) * WSTR + (t & 15)] = Wu_s[sb + t];     // shifted: 16x16 blocks
      sG[(t >> 4) * WSTR + (t & 15)] = Wg_s[sb + t];
      sD[(t >> 4) * WSTR + (t & 15)] = Wd_s[sb + t];
      if (t < 4)  { ebu[t] = bu_e[i * 4 + t];  ebg[t] = bg_e[i * 4 + t]; }
      if (t < 16) { sbu[t] = bu_s[i * 16 + t]; sbg[t] = bg_s[i * 16 + t]; }
    }
    __syncthreads();                           // weights visible to all 8 waves

    ffn_pair<1>(Xa, Xb, sca, scb, lane, eU, ebu, eG, ebg, eD);   // extract
    ffn_pair<4>(Xa, Xb, sca, scb, lane, sU, sbu, sG, sbg, sD);   // shifted

    __syncthreads();                           // all reads done before re-staging

    carry_pair(Xa, Xb, sca, scb, lane, cw);    // no shared-weight reads
  }

  float* o = out + (size_t)pair * 512;
#pragma unroll
  for (int r = 0; r < 8; ++r) {
    o[(r0 + r) * 16 + n]       = Xa[r];
    o[256 + (r0 + r) * 16 + n] = Xb[r];
  }
}

extern "C" void kernel_launch(void* const* d_in, const int* in_sizes, int n_in,
                              void* d_out, int out_size, void* d_ws, size_t ws_size,
                              hipStream_t stream) {
  (void)n_in; (void)d_ws; (void)ws_size; (void)out_size;
  const float* x    = (const float*)d_in[0];
  const float* Wu0  = (const float*)d_in[1];
  const float* bu0  = (const float*)d_in[2];
  const float* Wg0  = (const float*)d_in[3];
  const float* bg0  = (const float*)d_in[4];
  const float* Wd0  = (const float*)d_in[5];
  const float* Wu_e = (const float*)d_in[6];
  const float* bu_e = (const float*)d_in[7];
  const float* Wg_e = (const float*)d_in[8];
  const float* bg_e = (const float*)d_in[9];
  const float* Wd_e = (const float*)d_in[10];
  const float* Wu_s = (const float*)d_in[11];
  const float* bu_s = (const float*)d_in[12];
  const float* Wg_s = (const float*)d_in[13];
  const float* bg_s = (const float*)d_in[14];
  const float* Wd_s = (const float*)d_in[15];
  const float* Wu_c = (const float*)d_in[16];
  const float* bu_c = (const float*)d_in[17];
  const float* Wg_c = (const float*)d_in[18];
  const float* bg_c = (const float*)d_in[19];
  const float* Wd_c = (const float*)d_in[20];

  const int circuits = in_sizes[0] / 128;      // B  (P*D = 128 floats each)
  const int npairs   = circuits / 4;           // 4 circuits (2 tiles) per wave
  const int blocks   = (npairs + 7) / 8;       // 8 waves per 256-thread block

  pure_mul32_kernel<<<blocks, 256, 0, stream>>>(
      x, Wu0, bu0, Wg0, bg0, Wd0,
      Wu_e, bu_e, Wg_e, bg_e, Wd_e,
      Wu_s, bu_s, Wg_s, bg_s, Wd_s,
      Wu_c, bu_c, Wg_c, bg_c, Wd_c,
      (float*)d_out, npairs);
}